// SpectralConv4d_49435073577202
// MI455X (gfx1250) — compile-verified
//
#include <hip/hip_runtime.h>
#include <hip/hip_bf16.h>

// SpectralConv4d on gfx1250.
// Pipeline (per batch b): 4 forward partial-DFT stages (WMMA f32 16x16x4),
// per-mode channel mix (streaming, weights = HBM-bound 1.27GiB read),
// 3 inverse DFT stages + 1 c2r stage (WMMA), all staged through LDS.
// All intermediates planar (separate re/im float planes) for coalescing.
// Twiddle (DFT B-matrix) values are computed once per block into an LDS
// table so the steady-state loop is loads + v_wmma only (no sincos remat).

typedef __attribute__((ext_vector_type(2))) float v2f;
typedef __attribute__((ext_vector_type(8))) float v8f;

#define PI2_OVER_32 (6.28318530717958647692f / 32.0f)

static __device__ __forceinline__ v8f wmma_f32(v2f a, v2f b, v8f c) {
  // (neg_a, A, neg_b, B, c_mod, C, reuse_a, reuse_b)
  return __builtin_amdgcn_wmma_f32_16x16x4_f32(false, a, false, b, (short)0, c,
                                               false, false);
}

// Generic partial-DFT stage.
// Input viewed as [A=32, Q, M, PLEN] (floats, per plane), contracts Q.
// Output [A, M, PLEN, NM].  A-matrix = data rows (P dim), B = DFT matrix.
// MODE: 0 = forward, real input (stage S1)
//       1 = forward, complex input (S2,S3,S4)
//       2 = inverse, complex (I1,I2,I3), scale 1/32, freq map on q
//       3 = complex->real (I4), scale (1 or 2)/32
template <int Q, int NM, int PLEN, int MODE>
__global__ __launch_bounds__(256) void dft_step(
    const float* __restrict__ inRe, const float* __restrict__ inIm,
    float* __restrict__ outRe, float* __restrict__ outIm, int M, int nTasks) {
  constexpr int CH = Q / 4;              // K-chunks of 4
  constexpr int TILES = (NM + 15) / 16;  // B/N tiles
  constexpr int PG = (PLEN + 15) / 16;   // P groups per row-block
  constexpr int QP = Q + 2;              // LDS row stride (even, bank-safe)
  constexpr int INPLANES = (MODE == 0) ? 1 : 2;
  constexpr int WAVES = 8;
  constexpr int TPW = 4;  // tasks per wave

  // Staging buffers (per wave) + twiddle table (per block, read-only).
  __shared__ float lds[WAVES * INPLANES * 16 * QP];
  __shared__ float ldsB[2 * CH * TILES * 64];  // re table then im table

  const int tid = threadIdx.x;
  const int lane = tid & 31;
  const int wslot = tid >> 5;
  const int wid = blockIdx.x * WAVES + wslot;
  const int nLoc = lane & 15;  // N (B cols) / M (A rows) lane coordinate
  const int hq = lane >> 4;    // K half-select

  float* ldsRe = &lds[wslot * INPLANES * 16 * QP];
  float* ldsIm = ldsRe + 16 * QP;
  float* ldsBre = &ldsB[0];
  float* ldsBim = &ldsB[CH * TILES * 64];

  // ---- Build DFT B-matrix twiddle table once per block ----
  // For target lane L, chunk c, tile t the wave needs the v2f pair
  //   ( B[4c + 2*(L>>4) + 0][16t + (L&15)], B[... + 1][...] )
  // stored at ldsB[((c*TILES + t)*32 + L)*2 + j].
  for (int idx = tid; idx < CH * TILES * 32; idx += 256) {
    const int L = idx & 31;
    const int ct = idx >> 5;
    const int t = ct % TILES;
    const int c = ct / TILES;
    const int Lh = L >> 4;
    const int Ln = L & 15;
#pragma unroll
    for (int j = 0; j < 2; ++j) {
      const int q = 4 * c + 2 * Lh + j;
      const int n = 16 * t + Ln;
      float s, sn, cs, re, im;
      if constexpr (MODE == 0 || MODE == 1) {
        // forward: e^{-2 pi i f(n) q / 32}; f maps stored mode -> frequency
        const int f = (NM == 24) ? (n < 12 ? n : n + 8) : n;
        s = PI2_OVER_32 * (float)(f * q);
        __sincosf(s, &sn, &cs);
        re = cs; im = -sn;
      } else if constexpr (MODE == 2) {
        // inverse: (1/32) e^{+2 pi i f(q) n / 32}; q is the mode index
        const int f = (q < 12 ? q : q + 8);
        s = PI2_OVER_32 * (float)(f * n);
        __sincosf(s, &sn, &cs);
        re = cs * (1.0f / 32.0f); im = sn * (1.0f / 32.0f);
      } else {
        // c2r: out[t] = (1/32)[ReX0 + sum_k 2(ReXk cos - ImXk sin)]
        const float al = (q == 0 ? 1.0f : 2.0f) * (1.0f / 32.0f);
        s = PI2_OVER_32 * (float)(q * n);
        __sincosf(s, &sn, &cs);
        re = al * cs; im = -al * sn;
      }
      ldsBre[idx * 2 + j] = re;
      ldsBim[idx * 2 + j] = im;
    }
  }
  __syncthreads();  // once per block; table is read-only afterwards

  const int qStride = M * PLEN;

  for (int it = 0; it < TPW; ++it) {
    const int task = wid * TPW + it;  // wave-uniform
    if (task >= nTasks) break;
    const int g = task % PG;
    const int o = task / PG;
    const int a = o / M;
    const int m = o - a * M;
    const int inBase = a * (Q * qStride) + m * PLEN + g * 16;
    const int outBase = o * (PLEN * NM);
    int pv = PLEN - g * 16; if (pv > 16) pv = 16;

    // ---- Stage 16 rows x Q values per plane into LDS (coalesced 64B runs) ----
#pragma unroll
    for (int qq = 0; qq < Q / 2; ++qq) {
      const int q = 2 * qq + hq;
      const int p = nLoc;
      const bool v = (p < pv);
      const int idx = inBase + q * qStride + p;
      ldsRe[p * QP + q] = v ? inRe[idx] : 0.0f;
      if constexpr (INPLANES == 2) ldsIm[p * QP + q] = v ? inIm[idx] : 0.0f;
    }
    __asm__ volatile("s_wait_dscnt 0" ::: "memory");

    v8f accRe[TILES] = {};
    v8f accIm[TILES] = {};

    // ---- K loop: chained v_wmma_f32_16x16x4_f32 ----
#pragma unroll
    for (int c = 0; c < CH; ++c) {
      const int qb = 4 * c + 2 * hq;  // even -> 8B aligned LDS b64 read
      const v2f aRe = *(const v2f*)&ldsRe[nLoc * QP + qb];
      if constexpr (MODE == 0) {
#pragma unroll
        for (int t = 0; t < TILES; ++t) {
          const int bi = ((c * TILES + t) * 32 + lane) * 2;
          const v2f bre = *(const v2f*)&ldsBre[bi];
          const v2f bim = *(const v2f*)&ldsBim[bi];
          accRe[t] = wmma_f32(aRe, bre, accRe[t]);
          accIm[t] = wmma_f32(aRe, bim, accIm[t]);
        }
      } else {
        const v2f aIm = *(const v2f*)&ldsIm[nLoc * QP + qb];
        if constexpr (MODE == 3) {
#pragma unroll
          for (int t = 0; t < TILES; ++t) {
            const int bi = ((c * TILES + t) * 32 + lane) * 2;
            const v2f bre = *(const v2f*)&ldsBre[bi];
            const v2f bim = *(const v2f*)&ldsBim[bi];
            accRe[t] = wmma_f32(aRe, bre, accRe[t]);
            accRe[t] = wmma_f32(aIm, bim, accRe[t]);
          }
        } else {
          const v2f aImN = -aIm;  // f32 WMMA has no A-neg; do it in VALU
#pragma unroll
          for (int t = 0; t < TILES; ++t) {
            const int bi = ((c * TILES + t) * 32 + lane) * 2;
            const v2f bre = *(const v2f*)&ldsBre[bi];
            const v2f bim = *(const v2f*)&ldsBim[bi];
            accRe[t] = wmma_f32(aRe, bre, accRe[t]);
            accRe[t] = wmma_f32(aImN, bim, accRe[t]);
            accIm[t] = wmma_f32(aRe, bim, accIm[t]);
            accIm[t] = wmma_f32(aIm, bre, accIm[t]);
          }
        }
      }
    }

    // ---- Store D: VGPR r -> row (r + 8*hq), lane -> mode n; coalesced runs ----
#pragma unroll
    for (int t = 0; t < TILES; ++t) {
#pragma unroll
      for (int r = 0; r < 8; ++r) {
        const int row = r + 8 * hq;
        const int n = 16 * t + nLoc;
        if (row < pv && n < NM) {
          const int oidx = outBase + (g * 16 + row) * NM + n;
          outRe[oidx] = accRe[t][r];
          if constexpr (MODE != 3) outIm[oidx] = accIm[t][r];
        }
      }
    }
  }
}

// Per-mode channel mix: Y[co,mode] = sum_ci X[ci,mode] * W[corner,ci,co,mode]
// X/Y layout [c,24,24,24,12] planar; thread order follows the weight stream
// (HBM-bound 1.27GiB, perfectly coalesced 8B reads).
__global__ __launch_bounds__(256) void mix_kernel(
    const float* __restrict__ Xre, const float* __restrict__ Xim,
    const float* __restrict__ W, float* __restrict__ Yre,
    float* __restrict__ Yim) {
  const int m = blockIdx.x * 256 + threadIdx.x;  // 0..20735 flat (m1,m2,m3,m4)
  const int corner = blockIdx.y;                 // 0..7
  const int cog = blockIdx.z;                    // co group of 8

  const int m4 = m % 12; int r = m / 12;
  const int m3 = r % 12; r /= 12;
  const int m2 = r % 12;
  const int m1 = r / 12;

  const int S1v[8] = {0, 1, 0, 0, 1, 1, 0, 1};
  const int S2v[8] = {0, 0, 1, 0, 1, 0, 1, 1};
  const int S3v[8] = {0, 0, 0, 1, 0, 1, 1, 1};
  const int k1 = S1v[corner] * 12 + m1;
  const int k2 = S2v[corner] * 12 + m2;
  const int k3 = S3v[corner] * 12 + m3;
  const int xoff = ((k1 * 24 + k2) * 24 + k3) * 12 + m4;

  float accr[8], acci[8];
#pragma unroll
  for (int j = 0; j < 8; ++j) { accr[j] = 0.f; acci[j] = 0.f; }

  for (int ci = 0; ci < 32; ++ci) {
    const float xr = Xre[ci * 165888 + xoff];
    const float xi = Xim[ci * 165888 + xoff];
    const float* w =
        W + ((size_t)((corner * 32 + ci) * 32 + cog * 8) * 20736 + (size_t)m) * 2;
#pragma unroll
    for (int j = 0; j < 8; ++j) {
      const v2f wv = *(const v2f*)w;  // 8B aligned: offset is even
      accr[j] += xr * wv.x - xi * wv.y;
      acci[j] += xr * wv.y + xi * wv.x;
      w += 20736 * 2;
    }
  }
#pragma unroll
  for (int j = 0; j < 8; ++j) {
    const int co = cog * 8 + j;
    Yre[co * 165888 + xoff] = accr[j];
    Yim[co * 165888 + xoff] = acci[j];
  }
}

static inline dim3 gridFor(int nTasks) {
  return dim3((unsigned)((nTasks + 8 * 4 - 1) / (8 * 4)));
}

extern "C" void kernel_launch(void* const* d_in, const int* in_sizes, int n_in,
                              void* d_out, int out_size, void* d_ws,
                              size_t ws_size, hipStream_t stream) {
  const float* x = (const float*)d_in[0];  // [2,32,32,32,32,32]
  const float* w = (const float*)d_in[1];  // [8,32,32,12,12,12,12,2]
  float* out = (float*)d_out;              // [2,32,32,32,32,32]

  // workspace: two ping-pong regions, peak 336 MiB
  const size_t bufAbytes = 201326592;  // holds T1/T3/Y/U2 (per-b)
  const size_t bufBbytes = 150994944;  // holds T2/X1/U1/U3 (per-b)
  if (ws_size < bufAbytes + bufBbytes) return;
  float* bufA = (float*)d_ws;
  float* bufB = (float*)((char*)d_ws + bufAbytes);

  const size_t xb = 33554432;  // per-b elements of x / out

  for (int b = 0; b < 2; ++b) {
    const float* xin = x + (size_t)b * xb;
    float* outb = out + (size_t)b * xb;

    // per-b plane sizes (floats)
    float* T1r = bufA; float* T1i = bufA + 25165824;
    float* T2r = bufB; float* T2i = bufB + 18874368;
    float* T3r = bufA; float* T3i = bufA + 14155776;
    float* X1r = bufB; float* X1i = bufB + 5308416;
    float* Yr  = bufA; float* Yi  = bufA + 5308416;
    float* U1r = bufB; float* U1i = bufB + 7077888;
    float* U2r = bufA; float* U2i = bufA + 9437184;
    float* U3r = bufB; float* U3i = bufB + 12582912;

    // S1: x[ci,x,y,z,t] --contract x--> T1[ci,y,z,t,k1]   (real in)
    { int M = 1024, nt = 32 * M * 2;
      dft_step<32, 24, 32, 0><<<gridFor(nt), 256, 0, stream>>>(
          xin, nullptr, T1r, T1i, M, nt); }
    // S2: T1 --contract y--> T2[ci,z,t,k1,k2]
    { int M = 1024, nt = 32 * M * 2;
      dft_step<32, 24, 24, 1><<<gridFor(nt), 256, 0, stream>>>(
          T1r, T1i, T2r, T2i, M, nt); }
    // S3: T2 --contract z--> T3[ci,t,k1,k2,k3]
    { int M = 768, nt = 32 * M * 2;
      dft_step<32, 24, 24, 1><<<gridFor(nt), 256, 0, stream>>>(
          T2r, T2i, T3r, T3i, M, nt); }
    // S4: T3 --contract t--> X1[ci,k1,k2,k3,k4]
    { int M = 576, nt = 32 * M * 2;
      dft_step<32, 12, 24, 1><<<gridFor(nt), 256, 0, stream>>>(
          T3r, T3i, X1r, X1i, M, nt); }
    // Mix: X1 -> Y[co,k1,k2,k3,k4]
    mix_kernel<<<dim3(81, 8, 4), 256, 0, stream>>>(X1r, X1i, w, Yr, Yi);
    // I1: Y --contract k1--> U1[co,k2,k3,k4,x]
    { int M = 576, nt = 32 * M * 1;
      dft_step<24, 32, 12, 2><<<gridFor(nt), 256, 0, stream>>>(
          Yr, Yi, U1r, U1i, M, nt); }
    // I2: U1 --contract k2--> U2[co,k3,k4,x,y]
    { int M = 288, nt = 32 * M * 2;
      dft_step<24, 32, 32, 2><<<gridFor(nt), 256, 0, stream>>>(
          U1r, U1i, U2r, U2i, M, nt); }
    // I3: U2 --contract k3--> U3[co,k4,x,y,z]
    { int M = 384, nt = 32 * M * 2;
      dft_step<24, 32, 32, 2><<<gridFor(nt), 256, 0, stream>>>(
          U2r, U2i, U3r, U3i, M, nt); }
    // I4: U3 --c2r over k4--> out[co,x,y,z,t]
    { int M = 1024, nt = 32 * M * 2;
      dft_step<12, 32, 32, 3><<<gridFor(nt), 256, 0, stream>>>(
          U3r, U3i, outb, nullptr, M, nt); }
  }
}